// Tersoff_60498909331527
// MI455X (gfx1250) — compile-verified
//
#include <hip/hip_runtime.h>
#include <math.h>

typedef __attribute__((ext_vector_type(2))) float v2f;
typedef __attribute__((ext_vector_type(8))) float v8f;

// ---- Tersoff Si parameters (Tersoff 1988), pre-folded ----
#define LAMBDA3   1.3258f
#define C2_P      23.4072113f        // C_P^2
#define D2_P      4.16853889f        // D_P^2
#define G_BASE    6.61545177f        // 1 + C2/D2   (GAMMA = 1)
#define N_P       22.956f
#define BETA_N    1.40879e-11f       // BETA_P ^ N_P
#define NEG_INV2N (-0.0217807914f)   // -1/(2*N_P)
#define LAMBDA2   1.3258f
#define B_P       95.373f
#define R_CUT     3.0f
#define D_CUT     0.2f
#define LAMBDA1   3.2394f
#define A_P       3264.7f
#define SIN_K     7.85398163f        // pi / (2*D_CUT)

__device__ __forceinline__ float fcut(float r) {
    // 1 below R-D, sinusoidal taper on [R-D, R+D], 0 above
    float s = 0.5f - 0.5f * __sinf(SIN_K * (r - R_CUT));
    float o = (r < R_CUT + D_CUT) ? s : 0.0f;
    return (r < R_CUT - D_CUT) ? 1.0f : o;
}

// One wave (32 lanes) = one atom i:
//   16 in-edges k->i  (M rows of WMMA tile), 16 out-edges i->j (N cols).
//   Dot-product matrix via V_WMMA_F32_16X16X4_F32 (K-dim = xyz + zero pad).
//   zeta elementwise in the C/D register layout; column sum = segment_sum over k.
// Note: cos(theta) appears only squared (COSTHETA0 == 0), so the sign of
// r_ik = -r[idx_ik] is irrelevant -> feed raw vectors to the WMMA.
__global__ __launch_bounds__(256) void tersoff_wmma_kernel(
    const float* __restrict__ r,        // [E,3]
    const int*   __restrict__ idx_ik,   // [T]
    const int*   __restrict__ idx_ij,   // [T]
    float*       __restrict__ out,      // [E]
    int N)                              // atoms
{
    const int lane = threadIdx.x & 31;
    const int wave = threadIdx.x >> 5;          // 8 waves / block
    const int atom = blockIdx.x * 8 + wave;
    if (atom >= N) return;                      // wave-uniform: EXEC stays all-1s

    const int  K  = 16;
    const long t0 = (long)atom * K * K;         // this atom's K*K triplet block
    const int  e  = lane & 15;                  // edge slot (duplicated in hi half)

    // Edge ids from the index arrays (block structure: idx_ik repeats every K
    // entries along rows; first row enumerates idx_ij).
    const int eIn  = idx_ik[t0 + (long)e * K];  // edge k->i
    const int eOut = idx_ij[t0 + e];            // edge i->j

    const float* pi = r + (long)eIn  * 3;
    const float* po = r + (long)eOut * 3;
    const float xi = pi[0], yi = pi[1], zi = pi[2];
    const float xo = po[0], yo = po[1], zo = po[2];

    // Norms via rsq: 1 transcendental each, reciprocal comes for free.
    const float d2ik = xi * xi + yi * yi + zi * zi;
    const float d2ij = xo * xo + yo * yo + zo * zo;
    const float ibik_own = __builtin_amdgcn_rsqf(d2ik);
    const float ibij     = __builtin_amdgcn_rsqf(d2ij);
    const float blik = d2ik * ibik_own;         // |r_ik|
    const float blij = d2ij * ibij;             // |r_ij|
    const float fcik_own = fcut(blik);          // per-k, hoisted out of inner loop

    // A (16x4 f32): lanes 0-15 -> (K0,K1)=(x,y); lanes 16-31 -> (K2,K3)=(z,0).
    // B (4x16 f32): same striping by symmetry.
    const bool hi = lane >= 16;
    v2f a, b;
    a.x = hi ? zi : xi;  a.y = hi ? 0.0f : yi;
    b.x = hi ? zo : xo;  b.y = hi ? 0.0f : yo;

    v8f acc = {};
    // acc[v] = dot(r_ik_raw[M], r_ij[N]),  M = v + 8*hi,  N = lane&15
    acc = __builtin_amdgcn_wmma_f32_16x16x4_f32(
        false, a, false, b, (short)0, acc, false, false);

    // Elementwise zeta over this lane's 8 (M,N) cells; accumulate column sum.
    float colsum = 0.0f;
#pragma unroll
    for (int v = 0; v < 8; ++v) {
        const int   m    = (hi ? 8 : 0) + v;
        const float bik  = __shfl(blik,     m, 32);   // |r_ik[M]|
        const float ibik = __shfl(ibik_own, m, 32);   // 1/|r_ik[M]|
        const float fcik = __shfl(fcik_own, m, 32);   // fcut(|r_ik[M]|)
        // cos^2(theta); sign of dot irrelevant. Clamp |cos|<=1 on the square.
        const float t   = acc[v] * (ibij * ibik);
        const float dc2 = fminf(t * t, 1.0f);
        const float g   = G_BASE - C2_P * __builtin_amdgcn_rcpf(D2_P + dc2);
        const float diff = LAMBDA3 * (blij - bik);    // (l3*(dij-dik))^3
        const float zexp = __expf(diff * diff * diff);
        colsum = fmaf(fcik * g, zexp, colsum);
    }
    // segment_sum over k: fold the other half-wave (M=8..15 vs M=0..7)
    colsum += __shfl_xor(colsum, 16, 32);
    // lanes 0-15 (dup in 16-31) hold zeta_ij for out-edge j = lane&15

    // Per-bond energy
    const float zeta = colsum;
    const float zn   = (zeta > 0.0f) ? __expf(N_P * __logf(zeta)) : 0.0f;
    const float b_ij = __expf(NEG_INV2N * __logf(fmaf(BETA_N, zn, 1.0f)));
    const float energy = fcut(blij) *
        (A_P * __expf(-LAMBDA1 * blij) - b_ij * B_P * __expf(-LAMBDA2 * blij)) * 0.5f;

    if (lane < 16) out[eOut] = energy;   // coalesced: eOut = atom*16 + lane
}

extern "C" void kernel_launch(void* const* d_in, const int* in_sizes, int n_in,
                              void* d_out, int out_size, void* d_ws, size_t ws_size,
                              hipStream_t stream) {
    const float* r      = (const float*)d_in[0];
    const int*   idx_ik = (const int*)d_in[1];
    const int*   idx_ij = (const int*)d_in[2];
    float*       out    = (float*)d_out;

    const int E = in_sizes[0] / 3;   // edges
    const int N = E / 16;            // atoms (K = 16 neighbors)

    dim3 block(256);                 // 8 wave32 waves = 8 atoms per block
    dim3 grid((N + 7) / 8);
    tersoff_wmma_kernel<<<grid, block, 0, stream>>>(r, idx_ik, idx_ij, out, N);
}